// DifferentiableAugmentation_20830591385954
// MI455X (gfx1250) — compile-verified
//
#include <hip/hip_runtime.h>
#include <cstdint>
#include <cstddef>

#define IMG_H 224
#define IMG_W 224
#define IMG_C 3
#define ROWF (IMG_W * IMG_C)   // 672 floats per row
#define ROWB (ROWF * 4)        // 2688 bytes per row
#define COPY_LANES (ROWB / 16) // 168 lanes x b128

// ---------------------------------------------------------------------------
// Stage 1: resolve the stochastic augmentation decisions (tiny scalar work).
// Writes per-stage {mode, param} to workspace:
//   mode 0 = identity, 1 = shear_x (param = mag), 2 = translate_x (param = pix)
// ---------------------------------------------------------------------------
__global__ void aug_setup_params(const float* __restrict__ sp_weights,
                                 const float* __restrict__ sp_probs,
                                 const float* __restrict__ sp_magnitudes,
                                 const float* __restrict__ u_gumbel,
                                 const float* __restrict__ u_logistic,
                                 int* __restrict__ modes,
                                 float* __restrict__ params) {
    if (threadIdx.x != 0 || blockIdx.x != 0) return;
    const float eps = 1e-6f;
    for (int i = 0; i < 2; ++i) {
        // gumbel-softmax hard -> argmax(logits + g); first index wins ties
        float z[2];
        for (int j = 0; j < 2; ++j) {
            float u = fminf(fmaxf(u_gumbel[i * 2 + j], eps), 1.0f - eps);
            float g = -logf(-logf(u));
            z[j] = sp_weights[i * 2 + j] + g;
        }
        int js = (z[1] > z[0]) ? 1 : 0;
        // relaxed bernoulli hard -> round(sigmoid(...)), RNE like jnp.round
        float p = fminf(fmaxf(sp_probs[i * 2 + js], eps), 1.0f - eps);
        float u = fminf(fmaxf(u_logistic[i * 2 + js], eps), 1.0f - eps);
        float l = logf(u) - log1pf(-u);
        float t = logf(p) - log1pf(-p) + l;
        float y = 1.0f / (1.0f + expf(-t));
        int hw = (int)rintf(y);
        int mode = 0;
        float prm = 0.0f;
        if (hw == 1) {
            float m = sp_magnitudes[i * 2 + js];
            if (js == 0) { mode = 1; prm = __fsub_rn(__fmul_rn(0.6f, m), 0.3f); }   // shear mag
            else         { mode = 2; prm = __fsub_rn(__fmul_rn(20.0f, m), 10.0f); } // translate px
        }
        modes[i] = mode;
        params[i] = prm;
    }
}

// ---------------------------------------------------------------------------
// Stage 2: one block per (batch, row). Async-DMA the 2688 B input row into
// LDS (ASYNCcnt path, gfx1250), then emit the fused two-stage horizontal warp
// as a single LDS gather with zero-fill, coalesced b96 output stores.
// ---------------------------------------------------------------------------
__global__ __launch_bounds__(IMG_W) void aug_warp_rows(
        const float* __restrict__ in, float* __restrict__ out,
        const int* __restrict__ modes, const float* __restrict__ params) {
    // 16B-aligned so every b128 async LDS write is naturally aligned.
    __shared__ __align__(16) float row[ROWF];
    __shared__ int s_m0, s_m1;
    __shared__ float s_p0, s_p1;

    const int tid = threadIdx.x;
    const int rowid = blockIdx.x;          // b * H + y
    const int y = rowid % IMG_H;
    const size_t base = (size_t)rowid * ROWF;

    if (tid == 0) {
        s_m0 = modes[0]; s_m1 = modes[1];
        s_p0 = params[0]; s_p1 = params[1];
    }

    // Async DMA the input row into LDS: 168 lanes x 16 bytes (global 16B-aligned:
    // row byte offset = rowid * 2688, 2688 % 16 == 0).
    if (tid < COPY_LANES) {
        unsigned long long gaddr =
            (unsigned long long)(uintptr_t)in + (unsigned long long)base * 4ull +
            (unsigned long long)tid * 16ull;
        // Low 32 bits of a generic pointer to LDS are the LDS byte offset.
        unsigned lds_off = (unsigned)(uintptr_t)(void*)row + (unsigned)tid * 16u;
        asm volatile("global_load_async_to_lds_b128 %0, %1, off"
                     :: "v"(lds_off), "v"(gaddr) : "memory");
    }
    asm volatile("s_wait_asynccnt 0" ::: "memory");
    __syncthreads();

    const int m0 = s_m0, m1 = s_m1;
    const float p0 = s_p0, p1 = s_p1;
    const float yf = (float)y;

    // Compose: output <- stage1(stage0(input)).  Outer gather is stage i=1,
    // its source index is then warped again by stage i=0.
    int xa = tid;
    int ok = 1;

    if (m1 == 1) {
        int t = (int)rintf(__fadd_rn((float)xa, __fmul_rn(p1, yf)));
        ok &= ((unsigned)t < (unsigned)IMG_W) ? 1 : 0;
        xa = t < 0 ? 0 : (t > IMG_W - 1 ? IMG_W - 1 : t);
    } else if (m1 == 2) {
        int t = (int)rintf(__fsub_rn((float)xa, p1));
        ok &= ((unsigned)t < (unsigned)IMG_W) ? 1 : 0;
        xa = t < 0 ? 0 : (t > IMG_W - 1 ? IMG_W - 1 : t);
    }

    if (m0 == 1) {
        int t = (int)rintf(__fadd_rn((float)xa, __fmul_rn(p0, yf)));
        ok &= ((unsigned)t < (unsigned)IMG_W) ? 1 : 0;
        xa = t < 0 ? 0 : (t > IMG_W - 1 ? IMG_W - 1 : t);
    } else if (m0 == 2) {
        int t = (int)rintf(__fsub_rn((float)xa, p0));
        ok &= ((unsigned)t < (unsigned)IMG_W) ? 1 : 0;
        xa = t < 0 ? 0 : (t > IMG_W - 1 ? IMG_W - 1 : t);
    }

    float r0 = 0.0f, r1 = 0.0f, r2 = 0.0f;
    if (ok) {
        r0 = row[xa * 3 + 0];
        r1 = row[xa * 3 + 1];
        r2 = row[xa * 3 + 2];
    }
    out[base + (size_t)tid * 3 + 0] = r0;
    out[base + (size_t)tid * 3 + 1] = r1;
    out[base + (size_t)tid * 3 + 2] = r2;
}

// ---------------------------------------------------------------------------
extern "C" void kernel_launch(void* const* d_in, const int* in_sizes, int n_in,
                              void* d_out, int out_size, void* d_ws, size_t ws_size,
                              hipStream_t stream) {
    const float* x  = (const float*)d_in[0];
    const float* sw = (const float*)d_in[1];
    const float* sp = (const float*)d_in[2];
    const float* sm = (const float*)d_in[3];
    const float* ug = (const float*)d_in[4];
    const float* ul = (const float*)d_in[5];
    float* out = (float*)d_out;

    int*   modes  = (int*)d_ws;                  // 2 ints
    float* params = (float*)((char*)d_ws + 8);   // 2 floats

    hipLaunchKernelGGL(aug_setup_params, dim3(1), dim3(32), 0, stream,
                       sw, sp, sm, ug, ul, modes, params);

    const int B = in_sizes[0] / (IMG_H * IMG_W * IMG_C);  // 256
    hipLaunchKernelGGL(aug_warp_rows, dim3(B * IMG_H), dim3(IMG_W), 0, stream,
                       x, out, modes, params);
}